// SinkhornUnmatched_87600152969841
// MI455X (gfx1250) — compile-verified
//
#include <hip/hip_runtime.h>
#include <math.h>

// CDNA5 / gfx1250: wave32, WMMA f32 16x16x4 (exact fp32 matrix path).
typedef float v2f __attribute__((ext_vector_type(2)));
typedef float v8f __attribute__((ext_vector_type(8)));

#define WAVES_PER_BLOCK 4
#define ROWS_PER_WAVE   32
#define ROWS_PER_BLOCK  (WAVES_PER_BLOCK * ROWS_PER_WAVE)   // 128
#define ASTRIDE         68   // activation row stride (floats): conflict-free b64 LDS reads

__device__ __forceinline__ v8f wmma4(v2f a, v2f b, v8f c) {
  // D = A(16x4,f32) * B(4x16,f32) + C(16x16,f32)
  return __builtin_amdgcn_wmma_f32_16x16x4_f32(
      /*neg_a=*/false, a, /*neg_b=*/false, b,
      /*c_mod=*/(short)0, c, /*reuse_a=*/false, /*reuse_b=*/false);
}

__global__ __launch_bounds__(ROWS_PER_BLOCK)
void sinkhorn_mlp_wmma_kernel(
    const float* __restrict__ margins,
    const float* __restrict__ W1, const float* __restrict__ b1,
    const float* __restrict__ W2, const float* __restrict__ b2,
    const float* __restrict__ W3, const float* __restrict__ b3,
    const float* __restrict__ W4, const float* __restrict__ b4,
    float* __restrict__ out, long long nrows)
{
  // Weights staged pair-packed: w[((k>>1)*N + n)*2 + (k&1)]  -> B operand = one b64 load.
  __shared__ __align__(16) float w1s[8  * 64];   // K padded 6->8 (zeros)
  __shared__ __align__(16) float w2s[64 * 32];
  __shared__ __align__(16) float w3s[32 * 16];
  __shared__ __align__(16) float w4s[16 * 16];   // N padded 9->16 (zeros)
  __shared__ __align__(16) float b1s[64], b2s[32], b3s[16], b4s[16];
  __shared__ __align__(16) float acts[WAVES_PER_BLOCK][ROWS_PER_WAVE][ASTRIDE];

  const int tid = threadIdx.x;

  // ---- stage weights/biases (block-wide, once) ----
  for (int i = tid; i < 8 * 64; i += ROWS_PER_BLOCK) {
    int k = i >> 6, n = i & 63;
    w1s[(((k >> 1) * 64) + n) * 2 + (k & 1)] = (k < 6) ? W1[k * 64 + n] : 0.0f;
  }
  for (int i = tid; i < 64 * 32; i += ROWS_PER_BLOCK) {
    int k = i >> 5, n = i & 31;
    w2s[(((k >> 1) * 32) + n) * 2 + (k & 1)] = W2[k * 32 + n];
  }
  for (int i = tid; i < 32 * 16; i += ROWS_PER_BLOCK) {
    int k = i >> 4, n = i & 15;
    w3s[(((k >> 1) * 16) + n) * 2 + (k & 1)] = W3[k * 16 + n];
  }
  for (int i = tid; i < 16 * 16; i += ROWS_PER_BLOCK) {
    int k = i >> 4, n = i & 15;
    w4s[(((k >> 1) * 16) + n) * 2 + (k & 1)] = (n < 9) ? W4[k * 9 + n] : 0.0f;
  }
  if (tid < 64) b1s[tid] = b1[tid];
  if (tid < 32) b2s[tid] = b2[tid];
  if (tid < 16) b3s[tid] = b3[tid];
  if (tid < 16) b4s[tid] = (tid < 9) ? b4[tid] : 0.0f;
  __syncthreads();

  const int wv   = tid >> 5;
  const int lane = tid & 31;
  const int l16  = lane & 15;
  const int hi   = lane >> 4;          // 0: lanes 0-15 (K 0,1), 1: lanes 16-31 (K 2,3)
  const long long rowBase = (long long)blockIdx.x * ROWS_PER_BLOCK
                          + (long long)wv * ROWS_PER_WAVE;
  float (*act)[ASTRIDE] = acts[wv];    // wave-private transpose buffer

  // ================= Layer 1: 6 -> 64 (K padded to 8) =================
  #pragma unroll
  for (int mt = 0; mt < 2; ++mt) {
    const float* mp = margins + (rowBase + mt * 16 + l16) * 6;
    v2f a0; a0.x = mp[hi * 2 + 0]; a0.y = mp[hi * 2 + 1];   // K = 0..3
    v2f a1 = {0.0f, 0.0f};                                   // K = 4..7 (6,7 zero)
    if (!hi) { a1.x = mp[4]; a1.y = mp[5]; }
    #pragma unroll
    for (int nt = 0; nt < 4; ++nt) {
      const int n = nt * 16 + l16;
      const float bb = b1s[n];
      v8f acc = {bb, bb, bb, bb, bb, bb, bb, bb};
      v2f bB0 = *(const v2f*)&w1s[(((hi * 2) >> 1) * 64 + n) * 2];
      acc = wmma4(a0, bB0, acc);
      v2f bB1 = *(const v2f*)&w1s[((((4 + hi * 2)) >> 1) * 64 + n) * 2];
      acc = wmma4(a1, bB1, acc);
      #pragma unroll
      for (int r = 0; r < 8; ++r)                      // ReLU on D layout, store H1
        act[mt * 16 + hi * 8 + r][n] = fmaxf(acc[r], 0.0f);
    }
  }

  // ================= Layer 2: 64 -> 32 =================
  v8f h2[2][2];
  #pragma unroll
  for (int mt = 0; mt < 2; ++mt) {
    #pragma unroll
    for (int nt = 0; nt < 2; ++nt) {
      const int n = nt * 16 + l16;
      const float bb = b2s[n];
      v8f acc = {bb, bb, bb, bb, bb, bb, bb, bb};
      #pragma unroll
      for (int ks = 0; ks < 16; ++ks) {
        const int kb = ks * 4 + hi * 2;
        v2f a = *(const v2f*)&act[mt * 16 + l16][kb];
        v2f b = *(const v2f*)&w2s[((kb >> 1) * 32 + n) * 2];
        acc = wmma4(a, b, acc);
      }
      h2[mt][nt] = acc;
    }
  }
  #pragma unroll
  for (int mt = 0; mt < 2; ++mt)
    #pragma unroll
    for (int nt = 0; nt < 2; ++nt)
      #pragma unroll
      for (int r = 0; r < 8; ++r)
        act[mt * 16 + hi * 8 + r][nt * 16 + l16] = fmaxf(h2[mt][nt][r], 0.0f);

  // ================= Layer 3: 32 -> 16 =================
  v8f h3[2];
  #pragma unroll
  for (int mt = 0; mt < 2; ++mt) {
    const float bb = b3s[l16];
    v8f acc = {bb, bb, bb, bb, bb, bb, bb, bb};
    #pragma unroll
    for (int ks = 0; ks < 8; ++ks) {
      const int kb = ks * 4 + hi * 2;
      v2f a = *(const v2f*)&act[mt * 16 + l16][kb];
      v2f b = *(const v2f*)&w3s[((kb >> 1) * 16 + l16) * 2];
      acc = wmma4(a, b, acc);
    }
    h3[mt] = acc;
  }
  #pragma unroll
  for (int mt = 0; mt < 2; ++mt)
    #pragma unroll
    for (int r = 0; r < 8; ++r)
      act[mt * 16 + hi * 8 + r][l16] = fmaxf(h3[mt][r], 0.0f);

  // ================= Layer 4: 16 -> 9 (N padded to 16), no ReLU =================
  v8f p4[2];
  #pragma unroll
  for (int mt = 0; mt < 2; ++mt) {
    const float bb = b4s[l16];
    v8f acc = {bb, bb, bb, bb, bb, bb, bb, bb};
    #pragma unroll
    for (int ks = 0; ks < 4; ++ks) {
      const int kb = ks * 4 + hi * 2;
      v2f a = *(const v2f*)&act[mt * 16 + l16][kb];
      v2f b = *(const v2f*)&w4s[((kb >> 1) * 16 + l16) * 2];
      acc = wmma4(a, b, acc);
    }
    p4[mt] = acc;
  }
  #pragma unroll
  for (int mt = 0; mt < 2; ++mt)
    #pragma unroll
    for (int r = 0; r < 8; ++r)
      act[mt * 16 + hi * 8 + r][l16] = p4[mt][r];

  // ================= Sinkhorn epilogue: one row per lane =================
  const long long row = rowBase + lane;
  float pars[9];
  #pragma unroll
  for (int j = 0; j < 9; ++j) pars[j] = act[lane][j];

  const float* mp = margins + row * 6;
  const float M0 = mp[0], M1 = mp[1], M2 = mp[2];
  const float F0 = mp[3], F1 = mp[4], F2 = mp[5];

  const float p0 = expf(pars[0]), p1 = expf(pars[1]);
  const float p2 = expf(pars[2]), p3 = expf(pars[3]);
  float A[3][3];
  A[0][0] = p3 * p0; A[0][1] = p3 * p1; A[0][2] = p3 * p2;   // TAU = |i-j|
  A[1][0] = p3 * p1; A[1][1] = p3 * p0; A[1][2] = p3 * p1;
  A[2][0] = p3 * p2; A[2][1] = p3 * p1; A[2][2] = p3 * p0;

  const float shm0 = 1.0f / (1.0f + expf(-pars[4]));
  const float shm1 = 1.0f / (1.0f + expf(-pars[5]));
  const float shf0 = 1.0f / (1.0f + expf(-pars[6]));
  const float shf1 = 1.0f / (1.0f + expf(-pars[7]));
  const float V    = expf(pars[8]);

  const float rM[3]   = {M0 * shm0, M1 * shm1, M2};                 // mucm0
  const float cF[3]   = {F0 * shf0, F1 * shf1, F2};                 // muc0f
  const float mum0[3] = {M0 * (1.0f - shm0), M1 * (1.0f - shm1), 0.0f};
  const float mu0f[3] = {F0 * (1.0f - shf0), F1 * (1.0f - shf1), 0.0f};

  const float EPS = 1e-10f;
  #pragma unroll
  for (int it = 0; it < 10; ++it) {
    #pragma unroll
    for (int i = 0; i < 3; ++i) {
      const float s = A[i][0] + A[i][1] + A[i][2];
      const float r = rM[i] / (s + EPS);
      A[i][0] *= r; A[i][1] *= r; A[i][2] *= r;
    }
    #pragma unroll
    for (int j = 0; j < 3; ++j) {
      const float s = A[0][j] + A[1][j] + A[2][j];
      const float r = cF[j] / (s + EPS);
      A[0][j] *= r; A[1][j] *= r; A[2][j] *= r;
    }
  }

  // mus (B,4,4) then V (B,)
  float4* o4 = (float4*)out;
  const long long ob = row * 4;
  o4[ob + 0] = make_float4(A[0][0], A[0][1], A[0][2], mum0[0]);
  o4[ob + 1] = make_float4(A[1][0], A[1][1], A[1][2], mum0[1]);
  o4[ob + 2] = make_float4(A[2][0], A[2][1], A[2][2], mum0[2]);
  o4[ob + 3] = make_float4(mu0f[0], mu0f[1], mu0f[2], 0.0f);
  out[nrows * 16 + row] = V;
}

extern "C" void kernel_launch(void* const* d_in, const int* in_sizes, int n_in,
                              void* d_out, int out_size, void* d_ws, size_t ws_size,
                              hipStream_t stream) {
  const float* margins = (const float*)d_in[0];
  const float* W1 = (const float*)d_in[1]; const float* b1 = (const float*)d_in[2];
  const float* W2 = (const float*)d_in[3]; const float* b2 = (const float*)d_in[4];
  const float* W3 = (const float*)d_in[5]; const float* b3 = (const float*)d_in[6];
  const float* W4 = (const float*)d_in[7]; const float* b4 = (const float*)d_in[8];
  const long long nrows = (long long)in_sizes[0] / 6;      // B = 2,097,152
  const int blocks = (int)(nrows / ROWS_PER_BLOCK);        // exact: 16384
  sinkhorn_mlp_wmma_kernel<<<blocks, ROWS_PER_BLOCK, 0, stream>>>(
      margins, W1, b1, W2, b2, W3, b3, W4, b4, (float*)d_out, nrows);
}